// FamModelMoE_13357348291021
// MI455X (gfx1250) — compile-verified
//
#include <hip/hip_runtime.h>
#include <hip/hip_bf16.h>

typedef __attribute__((ext_vector_type(2))) float v2f;
typedef __attribute__((ext_vector_type(8))) float v8f;

#define NTOK 16384
#define DK   1024
#define CCL  32
#define HH   128
#define FPC  64
#define FT   2048     // C * FPC
#define KC   512      // K-chunk staged in LDS
#define XS_STRIDE 514 // 512 + 2 pad (even, avoids 64-bank conflicts)
#define HS_STRIDE 130 // 128 + 2 pad

// ---------------- kernel 1: init selected-clan scalar ----------------
__global__ void famoe_init_kernel(int* sel) {
    if (threadIdx.x == 0 && blockIdx.x == 0) *sel = 0x7fffffff;
}

// ---------------- kernel 2: per-token argmax routing ----------------
__global__ __launch_bounds__(256) void famoe_route_kernel(
    const float* __restrict__ xc, int* __restrict__ clan, int* __restrict__ sel) {
    __shared__ int smin[256];
    int n = blockIdx.x * 256 + threadIdx.x;
    const float* row = xc + (size_t)n * CCL;
    float best = row[0];
    int bi = 0;
    #pragma unroll
    for (int c = 1; c < CCL; ++c) {
        float v = row[c];
        if (v > best) { best = v; bi = c; }   // first-max == jnp.argmax
    }
    clan[n] = bi;
    smin[threadIdx.x] = bi;
    __syncthreads();
    for (int s = 128; s > 0; s >>= 1) {
        if (threadIdx.x < s) {
            int o = smin[threadIdx.x + s];
            if (o < smin[threadIdx.x]) smin[threadIdx.x] = o;
        }
        __syncthreads();
    }
    if (threadIdx.x == 0) atomicMin(sel, smin[0]);
}

// ---------------- kernel 3: fused expert (WMMA fp32) ----------------
__global__ __launch_bounds__(256) void famoe_main_kernel(
    const float* __restrict__ x,   const float* __restrict__ W1,
    const float* __restrict__ b1,  const float* __restrict__ gamma,
    const float* __restrict__ beta,const float* __restrict__ W2,
    const float* __restrict__ b2,  const int* __restrict__ clan,
    const int* __restrict__ selp,  float* __restrict__ out) {

    __shared__ float xs[16 * XS_STRIDE];     // staged x tile (16 x 512)
    __shared__ float hs[16 * HS_STRIDE];     // hidden tile (16 x 128)
    __shared__ float red_s[16 * 16];
    __shared__ float red_q[16 * 16];
    __shared__ float mu_row[16], rs_row[16];
    __shared__ int   mask_row[16];

    const int tid  = threadIdx.x;
    const int lane = tid & 31;
    const int wave = tid >> 5;     // 0..7
    const int hl   = lane >> 4;    // 0/1 (half-wave)
    const int lr   = lane & 15;
    const int m0   = blockIdx.x * 16;

    const int sel = *selp;
    const float* w1 = W1 + (size_t)sel * DK * HH;

    // -------- GEMM1: h[16x128] = x[16x1024] @ W1[1024x128] --------
    v8f acc = {};
    const int n0 = wave * 16;
    for (int kc = 0; kc < DK; kc += KC) {
        // stage 16 x 512 x-tile (2048 float4's, 8 per thread, coalesced)
        for (int i = tid; i < 16 * (KC / 4); i += 256) {
            int r  = i >> 7;            // 128 float4 per row
            int c4 = i & 127;
            float4 t = *(const float4*)(x + (size_t)(m0 + r) * DK + kc + c4 * 4);
            float* dst = &xs[r * XS_STRIDE + c4 * 4];
            dst[0] = t.x; dst[1] = t.y; dst[2] = t.z; dst[3] = t.w;
        }
        __syncthreads();
        #pragma unroll 4
        for (int k = 0; k < KC; k += 4) {
            v2f a, b;
            a.x = xs[lr * XS_STRIDE + k + 2 * hl];
            a.y = xs[lr * XS_STRIDE + k + 2 * hl + 1];
            int kg = kc + k + 2 * hl;
            b.x = w1[(size_t)kg * HH + n0 + lr];
            b.y = w1[(size_t)(kg + 1) * HH + n0 + lr];
            acc = __builtin_amdgcn_wmma_f32_16x16x4_f32(
                false, a, false, b, (short)0, acc, false, false);
        }
        __syncthreads();
    }

    // write h + b1 to LDS (D layout: vgpr v -> row v+8*hl, col n0+lr)
    {
        const float* b1s = b1 + sel * HH;
        int c = n0 + lr;
        float bb = b1s[c];
        #pragma unroll
        for (int v = 0; v < 8; ++v)
            hs[(v + 8 * hl) * HS_STRIDE + c] = acc[v] + bb;
    }
    __syncthreads();

    // -------- LayerNorm(128) + ReLU (16 threads per row) --------
    {
        int r = tid >> 4, l16 = tid & 15;
        float s = 0.f, q = 0.f;
        #pragma unroll
        for (int j = 0; j < 8; ++j) {
            float v = hs[r * HS_STRIDE + l16 * 8 + j];
            s += v; q += v * v;
        }
        red_s[r * 16 + l16] = s;
        red_q[r * 16 + l16] = q;
    }
    __syncthreads();
    if (tid < 16) {
        float ts = 0.f, tq = 0.f;
        #pragma unroll
        for (int j = 0; j < 16; ++j) { ts += red_s[tid * 16 + j]; tq += red_q[tid * 16 + j]; }
        float mu  = ts * (1.0f / 128.0f);
        float var = tq * (1.0f / 128.0f) - mu * mu;
        mu_row[tid] = mu;
        rs_row[tid] = rsqrtf(var + 1e-5f);
        mask_row[tid] = (clan[m0 + tid] == sel) ? 1 : 0;
    }
    __syncthreads();
    {
        const float* g  = gamma + sel * HH;
        const float* be = beta  + sel * HH;
        int r = tid >> 4, l16 = tid & 15;
        float mu = mu_row[r], rs = rs_row[r];
        #pragma unroll
        for (int j = 0; j < 8; ++j) {
            int c = l16 * 8 + j;
            float v = (hs[r * HS_STRIDE + c] - mu) * rs * g[c] + be[c];
            hs[r * HS_STRIDE + c] = v > 0.f ? v : 0.f;
        }
    }
    __syncthreads();

    // -------- zero-fill the 1984 non-selected columns --------
    {
        float4 z = {0.f, 0.f, 0.f, 0.f};
        for (int i = tid; i < 16 * (FT / 4); i += 256) {
            int r  = i >> 9;           // 512 float4 per row
            int c4 = i & 511;
            if ((c4 >> 4) != sel)      // skip the 16-float4 selected block
                *(float4*)(out + (size_t)(m0 + r) * FT + c4 * 4) = z;
        }
    }

    // -------- GEMM2: out16x64 = relu_h[16x128] @ W2[128x64], masked --------
    if (wave < 4) {                     // wave-uniform: EXEC all-ones inside
        const float* w2 = W2 + (size_t)sel * HH * FPC;
        const int nb = wave * 16;
        v8f acc2 = {};
        #pragma unroll 4
        for (int k = 0; k < HH; k += 4) {
            v2f a, b;
            a.x = hs[lr * HS_STRIDE + k + 2 * hl];
            a.y = hs[lr * HS_STRIDE + k + 2 * hl + 1];
            b.x = w2[(k + 2 * hl) * FPC + nb + lr];
            b.y = w2[(k + 2 * hl + 1) * FPC + nb + lr];
            acc2 = __builtin_amdgcn_wmma_f32_16x16x4_f32(
                false, a, false, b, (short)0, acc2, false, false);
        }
        float bb = b2[sel * FPC + nb + lr];
        #pragma unroll
        for (int v = 0; v < 8; ++v) {
            int r = v + 8 * hl;
            float val = mask_row[r] ? (acc2[v] + bb) : 0.0f;
            out[(size_t)(m0 + r) * FT + sel * FPC + nb + lr] = val;
        }
    }
}

extern "C" void kernel_launch(void* const* d_in, const int* in_sizes, int n_in,
                              void* d_out, int out_size, void* d_ws, size_t ws_size,
                              hipStream_t stream) {
    const float* x     = (const float*)d_in[0];
    const float* x_c   = (const float*)d_in[1];
    const float* W1    = (const float*)d_in[2];
    const float* b1    = (const float*)d_in[3];
    const float* gamma = (const float*)d_in[4];
    const float* beta  = (const float*)d_in[5];
    const float* W2    = (const float*)d_in[6];
    const float* b2    = (const float*)d_in[7];
    float* out = (float*)d_out;

    int* sel  = (int*)d_ws;            // 1 int (64B slot)
    int* clan = (int*)d_ws + 16;       // NTOK ints

    famoe_init_kernel<<<1, 32, 0, stream>>>(sel);
    famoe_route_kernel<<<NTOK / 256, 256, 0, stream>>>(x_c, clan, sel);
    famoe_main_kernel<<<NTOK / 16, 256, 0, stream>>>(
        x, W1, b1, gamma, beta, W2, b2, clan, sel, out);
}